// SAGE_58136677319059
// MI455X (gfx1250) — compile-verified
//
#include <hip/hip_runtime.h>

// ---------------------------------------------------------------------------
// GraphSAGE 2-layer forward for MI455X (gfx1250, wave32).
//   layer: h' = relu(h @ W_self + scatter_mean(h[src]->dst) @ W_neigh + b)
//   out   = h2 @ W_out + b_out
// GEMMs via V_WMMA_F32_16X16X4_F32 (fp32-exact tensor path; ~15 GFLOP total,
// negligible vs the edge phase). k-outer / col-tile-inner loop nest keeps all
// 8 accumulators live (64 VGPRs) with no spills; A fragments loaded once per
// k-step; B fragments batch-loaded per k-step from pre-packed weights so every
// load is a single global_load_b64 with an immediate offset.
// Edge gather/scatter stays L2-resident (51.2MB << 192MB L2).
// ---------------------------------------------------------------------------

typedef float v2f __attribute__((ext_vector_type(2)));
typedef float v8f __attribute__((ext_vector_type(8)));

#define FEAT 128
#define NCLS 64

__global__ void zero_f4_kernel(float4* __restrict__ p, int n4) {
    int i = blockIdx.x * blockDim.x + threadIdx.x;
    if (i < n4) p[i] = make_float4(0.f, 0.f, 0.f, 0.f);
}

__global__ void deg_kernel(const int* __restrict__ dst, float* __restrict__ deg,
                           int nedges) {
    int e = blockIdx.x * blockDim.x + threadIdx.x;
    if (e < nedges) {
        __hip_atomic_fetch_add(&deg[dst[e]], 1.0f, __ATOMIC_RELAXED,
                               __HIP_MEMORY_SCOPE_AGENT);
    }
}

__global__ void invdeg_kernel(float* __restrict__ deg, int n) {
    int i = blockIdx.x * blockDim.x + threadIdx.x;
    if (i < n) deg[i] = 1.0f / fmaxf(deg[i], 1.0f);
}

// Repack W[K][N] row-major into B-fragment order:
//   Wp[((k>>1)*N + n)*2 + (k&1)] = W[k][n]
// so a 16x16x4 B fragment (pair of consecutive K rows at one column) is a
// single aligned 8-byte load.
__global__ void pack_w_kernel(const float* __restrict__ W, float* __restrict__ Wp,
                              int K, int N) {
    int i = blockIdx.x * blockDim.x + threadIdx.x;
    if (i >= K * N) return;
    int k = i / N;
    int n = i - k * N;
    Wp[(((k >> 1) * N + n) << 1) + (k & 1)] = W[i];
}

// One edge per 32 lanes; each lane moves 4 consecutive floats (float4 load +
// 4 agent-scope fp32 atomics -> global_atomic_add_f32, no return).
__global__ void scatter_add_kernel(const float* __restrict__ h,
                                   const int* __restrict__ src,
                                   const int* __restrict__ dst,
                                   float* __restrict__ msg, int nedges) {
    int gid = blockIdx.x * blockDim.x + threadIdx.x;
    int e = gid >> 5;
    if (e >= nedges) return;
    int f = (gid & 31) * 4;
    int s = src[e];
    int d = dst[e];
    float4 v = *(const float4*)(h + (size_t)s * FEAT + f);
    float* p = msg + (size_t)d * FEAT + f;
    __hip_atomic_fetch_add(p + 0, v.x, __ATOMIC_RELAXED, __HIP_MEMORY_SCOPE_AGENT);
    __hip_atomic_fetch_add(p + 1, v.y, __ATOMIC_RELAXED, __HIP_MEMORY_SCOPE_AGENT);
    __hip_atomic_fetch_add(p + 2, v.z, __ATOMIC_RELAXED, __HIP_MEMORY_SCOPE_AGENT);
    __hip_atomic_fetch_add(p + 3, v.w, __ATOMIC_RELAXED, __HIP_MEMORY_SCOPE_AGENT);
}

// Fused SAGE GEMM: out[16 rows x NCOLS] per wave, k-outer loop with all
// NCOLS/16 accumulators resident.
//   out = act(A @ Wself + (Msg * invdeg[row]) @ Wneigh + bias)
// WMMA fp32 16x16x4 fragment layout (ISA 7.12.2):
//   A: lane l holds row m=l&15; VGPR0/1 = K = k + 2*(l>>4) + {0,1}
//   B: lane l holds col n=l&15;  VGPR0/1 = K rows k + 2*(l>>4) + {0,1}
//   C/D: VGPR r, lane l -> M = r + 8*(l>>4), N = l&15
template <int HAS_NEIGH, int RELU, int K, int NCOLS>
__global__ void sage_gemm_kernel(const float* __restrict__ A,
                                 const float* __restrict__ Msg,
                                 const float* __restrict__ invdeg,
                                 const float* __restrict__ WselfP,
                                 const float* __restrict__ WneighP,
                                 const float* __restrict__ bias,
                                 float* __restrict__ out,
                                 int nrows) {
    constexpr int NT = NCOLS / 16;  // number of 16-col output tiles
    const int lane = threadIdx.x & 31;
    const int wave = threadIdx.x >> 5;
    const int tile = blockIdx.x * (blockDim.x >> 5) + wave;
    const int tiles = (nrows + 15) >> 4;
    if (tile >= tiles) return;  // wave-uniform: EXEC stays all-ones for WMMA

    const int row0 = tile << 4;
    const int m = lane & 15;      // A-row / B-col / D-col index in tile
    const int phalf = lane >> 4;  // which half of the K-quad this lane holds
    const int koff = phalf << 1;  // 0 or 2

    float invd = 0.0f;
    if (HAS_NEIGH) invd = invdeg[row0 + m];

    const float* arow = A + (size_t)(row0 + m) * K + koff;
    const float* mrow = HAS_NEIGH ? (Msg + (size_t)(row0 + m) * K + koff) : nullptr;
    // packed-B lane base; fragment (k, tile t) at + (k>>1)*NCOLS*2 + t*32
    const float* bsb = WselfP + (((size_t)phalf * NCOLS + m) << 1);
    const float* bnb = HAS_NEIGH ? (WneighP + (((size_t)phalf * NCOLS + m) << 1))
                                 : nullptr;

    v8f acc[NT];
#pragma unroll
    for (int t = 0; t < NT; ++t)
        acc[t] = (v8f){0.f, 0.f, 0.f, 0.f, 0.f, 0.f, 0.f, 0.f};

#pragma unroll 2
    for (int k = 0; k < K; k += 4) {
        // A fragments: one 8B load each (loaded once per k-step)
        v2f a_s = *(const v2f*)(arow + k);
        v2f a_m;
        if (HAS_NEIGH) {
            a_m = *(const v2f*)(mrow + k);
            a_m.x *= invd;
            a_m.y *= invd;
        }
        // Batch all B-fragment loads for this k-step (one clause of b64 loads)
        v2f bs[NT], bn[NT];
#pragma unroll
        for (int t = 0; t < NT; ++t) {
            bs[t] = *(const v2f*)(bsb + (k >> 1) * NCOLS * 2 + t * 32);
            if (HAS_NEIGH)
                bn[t] = *(const v2f*)(bnb + (k >> 1) * NCOLS * 2 + t * 32);
        }
        // Two interleaved WMMA chains; same-accumulator ops are NT apart,
        // covering the WMMA->WMMA RAW latency.
#pragma unroll
        for (int t = 0; t < NT; ++t)
            acc[t] = __builtin_amdgcn_wmma_f32_16x16x4_f32(
                false, a_s, false, bs[t], (short)0, acc[t], false, false);
        if (HAS_NEIGH) {
#pragma unroll
            for (int t = 0; t < NT; ++t)
                acc[t] = __builtin_amdgcn_wmma_f32_16x16x4_f32(
                    false, a_m, false, bn[t], (short)0, acc[t], false, false);
        }
    }

    const int rbase = row0 + (phalf << 3);
#pragma unroll
    for (int t = 0; t < NT; ++t) {
        const int n = t * 16 + m;
        const float bb = bias[n];
#pragma unroll
        for (int r = 0; r < 8; ++r) {
            float v = acc[t][r] + bb;
            if (RELU) v = fmaxf(v, 0.0f);
            out[(size_t)(rbase + r) * NCOLS + n] = v;
        }
    }
}

static inline int cdiv(int a, int b) { return (a + b - 1) / b; }

extern "C" void kernel_launch(void* const* d_in, const int* in_sizes, int n_in,
                              void* d_out, int out_size, void* d_ws, size_t ws_size,
                              hipStream_t stream) {
    const float* x    = (const float*)d_in[0];
    const float* Ws1  = (const float*)d_in[1];
    const float* Wn1  = (const float*)d_in[2];
    const float* b1   = (const float*)d_in[3];
    const float* Ws2  = (const float*)d_in[4];
    const float* Wn2  = (const float*)d_in[5];
    const float* b2   = (const float*)d_in[6];
    const float* Wout = (const float*)d_in[7];
    const float* bout = (const float*)d_in[8];
    const int* esrc   = (const int*)d_in[9];
    const int* edst   = (const int*)d_in[10];

    const int N = in_sizes[0] / FEAT;   // 100000
    const int E = in_sizes[9];          // 1600000

    // workspace: invdeg | packed weights x5 | msg | h1 | h2
    char* ws = (char*)d_ws;
    size_t ofs = 0;
    auto carve = [&](size_t nfloats) {
        float* p = (float*)(ws + ofs);
        ofs = (ofs + nfloats * 4 + 255) & ~(size_t)255;
        return p;
    };
    float* invdeg = carve((size_t)N);
    float* Ws1p   = carve(FEAT * FEAT);
    float* Wn1p   = carve(FEAT * FEAT);
    float* Ws2p   = carve(FEAT * FEAT);
    float* Wn2p   = carve(FEAT * FEAT);
    float* Woutp  = carve(FEAT * NCLS);
    float* msg    = carve((size_t)N * FEAT);
    float* h1     = carve((size_t)N * FEAT);
    float* h2     = carve((size_t)N * FEAT);

    const int T = 256;
    const int gemm_blocks = cdiv(cdiv(N, 16), T / 32);
    const int scat_blocks = cdiv(E * 32, T);
    const int zero_feat_b = cdiv(N * (FEAT / 4), T);

    // pack weights into WMMA B-fragment order (tiny, L2-resident)
    pack_w_kernel<<<cdiv(FEAT * FEAT, T), T, 0, stream>>>(Ws1,  Ws1p,  FEAT, FEAT);
    pack_w_kernel<<<cdiv(FEAT * FEAT, T), T, 0, stream>>>(Wn1,  Wn1p,  FEAT, FEAT);
    pack_w_kernel<<<cdiv(FEAT * FEAT, T), T, 0, stream>>>(Ws2,  Ws2p,  FEAT, FEAT);
    pack_w_kernel<<<cdiv(FEAT * FEAT, T), T, 0, stream>>>(Wn2,  Wn2p,  FEAT, FEAT);
    pack_w_kernel<<<cdiv(FEAT * NCLS, T), T, 0, stream>>>(Wout, Woutp, FEAT, NCLS);

    // degree -> inverse degree
    zero_f4_kernel<<<cdiv(N / 4, T), T, 0, stream>>>((float4*)invdeg, N / 4);
    deg_kernel<<<cdiv(E, T), T, 0, stream>>>(edst, invdeg, E);
    invdeg_kernel<<<cdiv(N, T), T, 0, stream>>>(invdeg, N);

    // layer 1
    zero_f4_kernel<<<zero_feat_b, T, 0, stream>>>((float4*)msg, N * (FEAT / 4));
    scatter_add_kernel<<<scat_blocks, T, 0, stream>>>(x, esrc, edst, msg, E);
    sage_gemm_kernel<1, 1, FEAT, FEAT><<<gemm_blocks, T, 0, stream>>>(
        x, msg, invdeg, Ws1p, Wn1p, b1, h1, N);

    // layer 2
    zero_f4_kernel<<<zero_feat_b, T, 0, stream>>>((float4*)msg, N * (FEAT / 4));
    scatter_add_kernel<<<scat_blocks, T, 0, stream>>>(h1, esrc, edst, msg, E);
    sage_gemm_kernel<1, 1, FEAT, FEAT><<<gemm_blocks, T, 0, stream>>>(
        h1, msg, invdeg, Ws2p, Wn2p, b2, h2, N);

    // classifier (no neighbor term, no relu)
    sage_gemm_kernel<0, 0, FEAT, NCLS><<<gemm_blocks, T, 0, stream>>>(
        h2, nullptr, nullptr, Woutp, nullptr, bout, (float*)d_out, N);
}